// RipsEcc_10728828306118
// MI455X (gfx1250) — compile-verified
//
#include <hip/hip_runtime.h>

#define BATCHES 16
#define NPTS    512
#define STEPS   32
#define NBINS   33          // buckets 0..32 (32 = "beyond last threshold", clamped)
#define SPLITS  4
#define WAVES_PER_WG    8
#define WAVES_PER_BATCH (SPLITS * WAVES_PER_WG)   // 32
#define TILES   (NPTS / 16)                        // 32
#define TILE_SLOTS (TILES * TILES)                 // 1024 (upper triangle used)

typedef float v2f __attribute__((ext_vector_type(2)));
typedef float v8f __attribute__((ext_vector_type(8)));

__global__ void ecc_init(float* out) {
    int i = blockIdx.x * blockDim.x + threadIdx.x;
    if (i < BATCHES * STEPS) out[i] = (float)NPTS;   // vertex term: all P vertices at every t
}

__device__ __forceinline__ int bucket_of(float ni, float nj, float g) {
    float d2 = ni + nj - 2.0f * g;
    d2 = fmaxf(d2, 0.0f);                            // guard cancellation
    float d = __builtin_amdgcn_sqrtf(d2);            // raw v_sqrt_f32: plenty for 2/31 buckets
    int bkt = (int)ceilf(d * 15.5f);                 // smallest k with d <= k*(2/31)
    bkt = bkt < 0 ? 0 : (bkt > 32 ? 32 : bkt);
    return bkt;
}

__global__ __launch_bounds__(256)
void ecc_main(const float* __restrict__ x, float* __restrict__ out) {
    __shared__ float    pts[NPTS * 4];               // (x,y,z,0) per point
    __shared__ float    norms[NPTS];                 // |x_p|^2
    __shared__ unsigned hist[WAVES_PER_WG][NBINS];   // per-wave histograms
    __shared__ unsigned histT[NBINS];

    const int tid  = threadIdx.x;
    const int b    = blockIdx.x >> 2;   // batch
    const int s    = blockIdx.x & 3;    // split within batch
    const int lane = tid & 31;

    // Wave index as a compiler-known scalar: tile loop runs in SGPRs, so the
    // triangle skip is s_cbranch (EXEC provably all-ones at every WMMA).
    const int wave = __builtin_amdgcn_readfirstlane(tid >> 5);

    for (int k = tid; k < WAVES_PER_WG * NBINS; k += 256)
        ((unsigned*)hist)[k] = 0u;

    // Stage this batch's points + squared norms into LDS.
    const float* xb = x + (size_t)b * NPTS * 3;
    for (int p = tid; p < NPTS; p += 256) {
        float cx = xb[p * 3 + 0];
        float cy = xb[p * 3 + 1];
        float cz = xb[p * 3 + 2];
        pts[p * 4 + 0] = cx;
        pts[p * 4 + 1] = cy;
        pts[p * 4 + 2] = cz;
        pts[p * 4 + 3] = 0.0f;                        // K=3 pad
        norms[p] = cx * cx + cy * cy + cz * cz;
    }
    __syncthreads();

    const int waveInBatch = s * WAVES_PER_WG + wave;  // scalar, 0..31
    const int half = lane >> 4;                       // 0: coords (x,y)  1: coords (z,0)
    const int off  = half * 2;
    const int lp   = lane & 15;
    const int mbase = half * 8;

    for (int t = waveInBatch; t < TILE_SLOTS; t += WAVES_PER_BATCH) {
        const int I = t >> 5;                         // scalar
        const int J = t & 31;                         // scalar
        if (J < I) continue;                          // scalar skip of lower triangle
        const int i0 = I * 16, j0 = J * 16;

        // A operand: 16x4 f32; lane L<16 holds (K=0,K=1) of row M=L, lane L+16 holds (K=2,K=3).
        // B operand: 4x16 f32; same per-lane pattern for column N=lane&15 (same-set Gram block).
        v2f av, bv;
        av.x = pts[(i0 + lp) * 4 + off];
        av.y = pts[(i0 + lp) * 4 + off + 1];
        bv.x = pts[(j0 + lp) * 4 + off];
        bv.y = pts[(j0 + lp) * 4 + off + 1];

        v8f acc = {0.f, 0.f, 0.f, 0.f, 0.f, 0.f, 0.f, 0.f};
        // G[m][n] = <x_{i0+m}, x_{j0+n}> for the 16x16 tile, one tensor op.
        acc = __builtin_amdgcn_wmma_f32_16x16x4_f32(
            /*neg_a=*/false, av, /*neg_b=*/false, bv,
            /*c_mod=*/(short)0, acc, /*reuse_a=*/false, /*reuse_b=*/false);

        // C/D layout: VGPR r, lanes 0-15 -> (M=r, N=lane); lanes 16-31 -> (M=r+8, N=lane-16).
        const int   j  = j0 + lp;
        const float nj = norms[j];

        // Hoist row norms (half-uniform LDS broadcast reads) out of any divergence.
        float ni[8];
        #pragma unroll
        for (int r = 0; r < 8; ++r) ni[r] = norms[i0 + mbase + r];

        if (I == J) {
            // Diagonal tile: per-lane guard keeps each edge counted once.
            #pragma unroll
            for (int r = 0; r < 8; ++r) {
                const int i = i0 + mbase + r;
                if (i < j) {
                    int bkt = bucket_of(ni[r], nj, acc[r]);
                    atomicAdd(&hist[wave][bkt], 1u);
                }
            }
        } else {
            // Off-diagonal tile (I < J): every (i,j) has i < j — no guards, no exec churn.
            #pragma unroll
            for (int r = 0; r < 8; ++r) {
                int bkt = bucket_of(ni[r], nj, acc[r]);
                atomicAdd(&hist[wave][bkt], 1u);
            }
        }
    }
    __syncthreads();

    // Reduce per-wave histograms, prefix-sum, commit float counts (exact: < 2^24).
    if (tid < NBINS) {
        unsigned sum = 0;
        #pragma unroll
        for (int w = 0; w < WAVES_PER_WG; ++w) sum += hist[w][tid];
        histT[tid] = sum;
    }
    __syncthreads();

    if (tid < STEPS) {
        unsigned prefix = 0;
        for (int m = 0; m <= tid; ++m) prefix += histT[m];
        atomicAdd(&out[b * STEPS + tid], -(float)prefix);  // ecc = P - n_edges
    }
}

extern "C" void kernel_launch(void* const* d_in, const int* in_sizes, int n_in,
                              void* d_out, int out_size, void* d_ws, size_t ws_size,
                              hipStream_t stream) {
    const float* x   = (const float*)d_in[0];   // [16, 512, 3] float32
    float*       out = (float*)d_out;           // [16, 32] float32

    ecc_init<<<dim3((BATCHES * STEPS + 255) / 256), dim3(256), 0, stream>>>(out);
    ecc_main<<<dim3(BATCHES * SPLITS), dim3(256), 0, stream>>>(x, out);
}